// HRRSelfAttention_74448963109043
// MI455X (gfx1250) — compile-verified
//
#include <hip/hip_runtime.h>

// HRR self-attention for MI455X (gfx1250, wave32).
// Pipeline: [WMMA-f32 GEMM qkv] -> [bind circ-conv] -> [3-phase causal scan]
//           -> [unbind circ-corr, in place] -> [WMMA-f32 GEMM out].
// fp32 kept end-to-end (K=1024 reductions; bf16 would be too lossy) using
// V_WMMA_F32_16X16X4_F32 (exact fp32 FMA on the matrix pipe).
// GEMM tiles are streamed global->LDS with CDNA5 async-to-LDS (ASYNCcnt),
// double-buffered so loads overlap the WMMA stream.

typedef __attribute__((ext_vector_type(2))) float v2f;
typedef __attribute__((ext_vector_type(8))) float v8f;

#define BB 4
#define SS 2048
#define MM 1024
#define HH 8
#define DD 128
#define RR (BB * SS)          // 8192 rows
#define TT 32                 // scan tiles over S
#define TILE_S (SS / TT)      // 64

// ---- CDNA5 async global->LDS (bypasses VGPRs, tracked by ASYNCcnt) --------
__device__ __forceinline__ void async_ld_b128(const void* lds_dst, const void* gsrc) {
    // LDS byte offset = low 32 bits of the flat shared-memory address.
    asm volatile("global_load_async_to_lds_b128 %0, %1, off"
                 :: "v"((unsigned int)(unsigned long long)(uintptr_t)lds_dst),
                    "v"((unsigned long long)(uintptr_t)gsrc)
                 : "memory");
}
__device__ __forceinline__ void wait_async0() {
    asm volatile("s_wait_asynccnt 0" ::: "memory");
}

// ---------------------------------------------------------------------------
// GEMM: Y[r,n] = sum_m X[r,m] * W[n,m]   (X:[R,Mk], W:[N,Mk], row-major)
// Block = 128 threads = 4 waves. Block tile 64x64; wave = 16 rows x 64 cols
// via 4 accumulators (A fragment reused x4). Double-buffered async LDS tiles.
//
// Fragment layouts (ISA 7.12.2, wave32):
//   A 16x4 f32 : lanes 0-15 -> row M=lane, K{0,1}; lanes 16-31 -> K{2,3}.
//   B 4x16 f32 : mirrored (lane -> column N, half-wave selects K pair).
//   C/D 16x16  : VGPR r -> M=r (lanes 0-15) / M=r+8 (lanes 16-31), N=lane&15.
// ---------------------------------------------------------------------------
__global__ __launch_bounds__(128)
void hrr_gemm_xWt(const float* __restrict__ X, const float* __restrict__ W,
                  float* __restrict__ Y, int Mk, int Ncols) {
    __shared__ float Xs[2][64][68];   // pitch 68: (row*68)%64 = row*4, conflict-free
    __shared__ float Ws[2][64][68];   // row base 272B = 17*16B -> b128-aligned

    const int tid     = threadIdx.x;
    const int lane    = tid & 31;
    const int wave    = tid >> 5;
    const int lanelo  = lane & 15;
    const int laneHi  = lane >> 4;
    const int rowBase = blockIdx.y * 64;
    const int colBase = blockIdx.x * 64;

    const int ldRow = (tid >> 4) << 3;      // this thread loads 8 consecutive rows
    const int ldCol = (tid & 15) * 4;       // float4 column within the K-chunk

    auto issue = [&](int buf, int kc) {
        const float* Xg = X + (size_t)(rowBase + ldRow) * Mk + kc * 64 + ldCol;
        const float* Wg = W + (size_t)(colBase + ldRow) * Mk + kc * 64 + ldCol;
        #pragma unroll
        for (int it = 0; it < 8; ++it) {
            async_ld_b128(&Xs[buf][ldRow + it][ldCol], Xg + (size_t)it * Mk);
            async_ld_b128(&Ws[buf][ldRow + it][ldCol], Wg + (size_t)it * Mk);
        }
    };

    v8f acc0 = {0,0,0,0,0,0,0,0};
    v8f acc1 = acc0, acc2 = acc0, acc3 = acc0;

    const int nChunks = Mk >> 6;            // Kc = 64
    issue(0, 0);
    for (int kc = 0; kc < nChunks; ++kc) {
        wait_async0();                      // this wave's LDS tile writes done
        __syncthreads();                    // all waves' writes visible; prev reads done
        if (kc + 1 < nChunks) issue((kc + 1) & 1, kc + 1);   // overlap with compute

        const float (*Xsb)[68] = Xs[kc & 1];
        const float (*Wsb)[68] = Ws[kc & 1];
        #pragma unroll 4
        for (int ks = 0; ks < 16; ++ks) {   // 16 k-steps of K=4
            const int k0 = ks * 4 + laneHi * 2;
            v2f a  = *(const v2f*)&Xsb[wave * 16 + lanelo][k0];
            v2f b0 = *(const v2f*)&Wsb[ 0 + lanelo][k0];
            v2f b1 = *(const v2f*)&Wsb[16 + lanelo][k0];
            v2f b2 = *(const v2f*)&Wsb[32 + lanelo][k0];
            v2f b3 = *(const v2f*)&Wsb[48 + lanelo][k0];
            acc0 = __builtin_amdgcn_wmma_f32_16x16x4_f32(false, a, false, b0, (short)0, acc0, false, false);
            acc1 = __builtin_amdgcn_wmma_f32_16x16x4_f32(false, a, false, b1, (short)0, acc1, false, false);
            acc2 = __builtin_amdgcn_wmma_f32_16x16x4_f32(false, a, false, b2, (short)0, acc2, false, false);
            acc3 = __builtin_amdgcn_wmma_f32_16x16x4_f32(false, a, false, b3, (short)0, acc3, false, false);
        }
        __syncthreads();                    // done reading buf before it is refilled
    }

    const int rowOut = rowBase + wave * 16 + laneHi * 8;
    #pragma unroll
    for (int r = 0; r < 8; ++r) {
        size_t base = (size_t)(rowOut + r) * Ncols + colBase + lanelo;
        Y[base +  0] = acc0[r];
        Y[base + 16] = acc1[r];
        Y[base + 32] = acc2[r];
        Y[base + 48] = acc3[r];
    }
}

// ---------------------------------------------------------------------------
// HRR bind: kv[bs,h,j] = sum_i k[i] * v[(j-i) mod D].  One block per (bs,h).
// ---------------------------------------------------------------------------
__global__ __launch_bounds__(DD)
void hrr_bind(const float* __restrict__ qkv, float* __restrict__ kv) {
    const int blk = blockIdx.x;             // bs*H + h
    const int h   = blk & (HH - 1);
    const int bs  = blk >> 3;
    const float* kp = qkv + (size_t)bs * (3 * MM) + MM     + h * DD;
    const float* vp = qkv + (size_t)bs * (3 * MM) + 2 * MM + h * DD;

    __shared__ float ks[DD];
    __shared__ float vs[DD];
    const int j = threadIdx.x;
    ks[j] = kp[j];
    vs[j] = vp[j];
    __syncthreads();

    float acc = 0.f;
    #pragma unroll 8
    for (int i = 0; i < DD; ++i)
        acc = __builtin_fmaf(ks[i], vs[(j - i) & (DD - 1)], acc);
    kv[(size_t)blk * DD + j] = acc;
}

// ---------------------------------------------------------------------------
// Causal cumsum over S, 3-phase scan for memory parallelism (1024 blocks).
// ---------------------------------------------------------------------------
__global__ __launch_bounds__(DD)
void hrr_scan_partial(const float* __restrict__ kv, float* __restrict__ part) {
    const int t  = blockIdx.x & (TT - 1);
    const int bh = blockIdx.x / TT;
    const int b  = bh >> 3, h = bh & (HH - 1);
    const int d  = threadIdx.x;
    const int s0 = t * TILE_S;
    float acc = 0.f;
    for (int s = 0; s < TILE_S; ++s)
        acc += kv[(((size_t)(b * SS + s0 + s)) * HH + h) * DD + d];
    part[((size_t)bh * TT + t) * DD + d] = acc;
}

__global__ __launch_bounds__(DD)
void hrr_scan_offsets(float* __restrict__ part) {
    const int bh = blockIdx.x;
    const int d  = threadIdx.x;
    float acc = 0.f;
    for (int t = 0; t < TT; ++t) {
        size_t idx = ((size_t)bh * TT + t) * DD + d;
        float v = part[idx];
        part[idx] = acc;                    // exclusive prefix
        acc += v;
    }
}

__global__ __launch_bounds__(DD)
void hrr_scan_apply(float* __restrict__ kv, const float* __restrict__ part) {
    const int t  = blockIdx.x & (TT - 1);
    const int bh = blockIdx.x / TT;
    const int b  = bh >> 3, h = bh & (HH - 1);
    const int d  = threadIdx.x;
    const int s0 = t * TILE_S;
    float acc = part[((size_t)bh * TT + t) * DD + d];
    for (int s = 0; s < TILE_S; ++s) {
        size_t idx = (((size_t)(b * SS + s0 + s)) * HH + h) * DD + d;
        acc += kv[idx];
        kv[idx] = acc;
    }
}

// ---------------------------------------------------------------------------
// HRR unbind: out[j] = sum_i cumkv[(i+j) mod D] * q[i], written in place.
// kv flat [B,S,H,D] aliases the [B*S, M] attention matrix (m = h*D + j).
// ---------------------------------------------------------------------------
__global__ __launch_bounds__(DD)
void hrr_unbind(const float* __restrict__ qkv, float* __restrict__ kv) {
    const int blk = blockIdx.x;
    const int h   = blk & (HH - 1);
    const int bs  = blk >> 3;
    const float* qp = qkv + (size_t)bs * (3 * MM) + h * DD;

    __shared__ float qs[DD];
    __shared__ float cs[DD];
    const int j = threadIdx.x;
    qs[j] = qp[j];
    cs[j] = kv[(size_t)blk * DD + j];
    __syncthreads();

    float acc = 0.f;
    #pragma unroll 8
    for (int i = 0; i < DD; ++i)
        acc = __builtin_fmaf(cs[(i + j) & (DD - 1)], qs[i], acc);
    kv[(size_t)blk * DD + j] = acc;         // reads already staged in LDS
}

// ---------------------------------------------------------------------------
extern "C" void kernel_launch(void* const* d_in, const int* in_sizes, int n_in,
                              void* d_out, int out_size, void* d_ws, size_t ws_size,
                              hipStream_t stream) {
    const float* x     = (const float*)d_in[0];   // [B,S,M]
    const float* W_qkv = (const float*)d_in[1];   // [3M,M]
    const float* W_o   = (const float*)d_in[2];   // [M,M]
    float* out = (float*)d_out;                   // [B,S,M]

    float* qkv  = (float*)d_ws;                   // R*3M floats (96 MB)
    float* kv   = qkv + (size_t)RR * 3 * MM;      // R*M  floats (32 MB)
    float* part = kv + (size_t)RR * MM;           // B*H*TT*D floats (512 KB)

    // 1) qkv = x @ W_qkv^T
    {
        dim3 grid((3 * MM) / 64, RR / 64);
        hrr_gemm_xWt<<<grid, 128, 0, stream>>>(x, W_qkv, qkv, MM, 3 * MM);
    }
    // 2) bind (circular convolution per head)
    hrr_bind<<<RR * HH, DD, 0, stream>>>(qkv, kv);
    // 3) causal cumulative sum over sequence (3-phase scan)
    hrr_scan_partial<<<BB * HH * TT, DD, 0, stream>>>(kv, part);
    hrr_scan_offsets<<<BB * HH, DD, 0, stream>>>(part);
    hrr_scan_apply<<<BB * HH * TT, DD, 0, stream>>>(kv, part);
    // 4) unbind (circular correlation with q), in place -> attention matrix
    hrr_unbind<<<RR * HH, DD, 0, stream>>>(qkv, kv);
    // 5) out = attn @ W_o^T
    {
        dim3 grid(MM / 64, RR / 64);
        hrr_gemm_xWt<<<grid, 128, 0, stream>>>(kv, W_o, out, MM, MM);
    }
}